// ChemicalNet_57750130262457
// MI455X (gfx1250) — compile-verified
//
#include <hip/hip_runtime.h>
#include <math.h>

#define NS   4
#define DE   128
#define HID  256
#define TM   16

typedef __attribute__((ext_vector_type(2))) float v2f;
typedef __attribute__((ext_vector_type(8))) float v8f;

__device__ __forceinline__ v8f wmma4(v2f a, v2f b, v8f c) {
    // D = A(16x4 f32) x B(4x16 f32) + C(16x16 f32)
    return __builtin_amdgcn_wmma_f32_16x16x4_f32(
        /*neg_a=*/false, a, /*neg_b=*/false, b,
        /*c_mod=*/(short)0, c, /*reuse_a=*/false, /*reuse_b=*/false);
}

__device__ __forceinline__ float silu(float x) {
    return x / (1.0f + __expf(-x));
}

// ---------------------------------------------------------------------------
// Pass 0: zero output, reset per-species counters
// ---------------------------------------------------------------------------
__global__ void cn_init_kernel(float* __restrict__ out, int n, int* __restrict__ counters) {
    int i = blockIdx.x * blockDim.x + threadIdx.x;
    if (i < n) out[i] = 0.0f;
    if (i < NS) counters[i] = 0;
}

// ---------------------------------------------------------------------------
// Pass 1: bucket atoms by species (atomic number -> species index)
// ---------------------------------------------------------------------------
__global__ void cn_bucket_kernel(const int* __restrict__ species, int n,
                                 int* __restrict__ counters, int* __restrict__ lists,
                                 int cap) {
    int i = blockIdx.x * blockDim.x + threadIdx.x;
    if (i >= n) return;
    int z = species[i];
    int idx = (z == 1) ? 0 : (z == 6) ? 1 : (z == 7) ? 2 : (z == 8) ? 3 : -1;
    if (idx >= 0) {
        int pos = atomicAdd(&counters[idx], 1);
        lists[idx * cap + pos] = i;
    }
}

// ---------------------------------------------------------------------------
// Pass 2: per-species 3-layer MLP on 16-atom tiles, WMMA f32 16x16x4
// ---------------------------------------------------------------------------
__global__ __launch_bounds__(256)
void cn_mlp_kernel(const float* __restrict__ emb,
                   const float* __restrict__ W1, const float* __restrict__ b1,
                   const float* __restrict__ W2, const float* __restrict__ b2,
                   const float* __restrict__ W3, const float* __restrict__ b3,
                   const int* __restrict__ counters, const int* __restrict__ lists,
                   int cap, float* __restrict__ out) {
    __shared__ int   ids[TM];
    __shared__ float Ash[TM][DE + 4];    // stride 132: (m*132+k)%64 = 4m+k -> conflict-free
    __shared__ float H1[TM][HID + 4];    // stride 260: same property
    __shared__ float H2[TM][HID + 4];
    __shared__ float red[TM][17];

    const int s    = blockIdx.y;
    const int cnt  = counters[s];
    const int base = blockIdx.x * TM;
    if (base >= cnt) return;             // block-uniform early exit

    const int tid   = threadIdx.x;
    const int wave  = tid >> 5;
    const int lane  = tid & 31;
    const int mrow  = lane & 15;         // A: M row / B: N col / D: N col
    const int khalf = lane >> 4;         // which K-half of the 16x4 fragment

    if (tid < TM) {
        int r = base + tid;
        ids[tid] = (r < cnt) ? lists[s * cap + r] : -1;
    }
    __syncthreads();

    // stage 16x128 embedding tile (float4-vectorized gather)
    for (int v = tid; v < TM * (DE / 4); v += blockDim.x) {
        int row = v >> 5;                // DE/4 = 32 float4 per row
        int c4  = v & 31;
        int id  = ids[row];
        float4 d = make_float4(0.f, 0.f, 0.f, 0.f);
        if (id >= 0) d = ((const float4*)(emb + (size_t)id * DE))[c4];
        Ash[row][c4 * 4 + 0] = d.x;  Ash[row][c4 * 4 + 1] = d.y;
        Ash[row][c4 * 4 + 2] = d.z;  Ash[row][c4 * 4 + 3] = d.w;
    }
    __syncthreads();

    // ----- Layer 1: [16x128] @ [128x256] + b1, SiLU -----
    {
        const float* Wp = W1 + (size_t)s * DE * HID;
        const float* bp = b1 + s * HID;
        for (int nt = wave * 2; nt < wave * 2 + 2; ++nt) {   // 8 waves x 2 = 16 n-tiles
            int ncol = nt * 16 + mrow;
            v8f acc = {};
            for (int k0 = 0; k0 < DE; k0 += 4) {
                int k = k0 + 2 * khalf;
                v2f a; a.x = Ash[mrow][k];               a.y = Ash[mrow][k + 1];
                v2f b; b.x = Wp[(size_t)k * HID + ncol]; b.y = Wp[(size_t)(k + 1) * HID + ncol];
                acc = wmma4(a, b, acc);
            }
            float bias = bp[ncol];
#pragma unroll
            for (int r = 0; r < 8; ++r) {
                int row = r + 8 * khalf;
                H1[row][ncol] = silu(acc[r] + bias);
            }
        }
    }
    __syncthreads();

    // ----- Layer 2: [16x256] @ [256x256] + b2, SiLU -----
    {
        const float* Wp = W2 + (size_t)s * HID * HID;
        const float* bp = b2 + s * HID;
        for (int nt = wave * 2; nt < wave * 2 + 2; ++nt) {
            int ncol = nt * 16 + mrow;
            v8f acc = {};
            for (int k0 = 0; k0 < HID; k0 += 4) {
                int k = k0 + 2 * khalf;
                v2f a; a.x = H1[mrow][k];                a.y = H1[mrow][k + 1];
                v2f b; b.x = Wp[(size_t)k * HID + ncol]; b.y = Wp[(size_t)(k + 1) * HID + ncol];
                acc = wmma4(a, b, acc);
            }
            float bias = bp[ncol];
#pragma unroll
            for (int r = 0; r < 8; ++r) {
                int row = r + 8 * khalf;
                H2[row][ncol] = silu(acc[r] + bias);
            }
        }
    }
    __syncthreads();

    // ----- Layer 3: [16x256] @ [256x1] + b3 (tiny; VALU + LDS reduce) -----
    {
        const float* Wp = W3 + s * HID;
        int row   = tid & 15;
        int chunk = tid >> 4;            // 16 chunks of 16 K each
        float p = 0.f;
#pragma unroll
        for (int j = 0; j < 16; ++j)
            p += H2[row][chunk * 16 + j] * Wp[chunk * 16 + j];
        red[row][chunk] = p;
        __syncthreads();
        if (tid < TM) {
            float sum = b3[s];
#pragma unroll
            for (int c = 0; c < 16; ++c) sum += red[tid][c];
            int id = ids[tid];
            if (id >= 0) out[id] = sum;
        }
    }
}

// ---------------------------------------------------------------------------
extern "C" void kernel_launch(void* const* d_in, const int* in_sizes, int n_in,
                              void* d_out, int out_size, void* d_ws, size_t ws_size,
                              hipStream_t stream) {
    const int*   species = (const int*)  d_in[0];
    const float* emb     = (const float*)d_in[1];
    const float* W1      = (const float*)d_in[2];
    const float* b1      = (const float*)d_in[3];
    const float* W2      = (const float*)d_in[4];
    const float* b2      = (const float*)d_in[5];
    const float* W3      = (const float*)d_in[6];
    const float* b3      = (const float*)d_in[7];
    float* out = (float*)d_out;
    const int n = in_sizes[0];

    int* counters = (int*)d_ws;
    int* lists    = counters + 8;        // 32B-aligned lists region, 4*n ints

    int blocks = (n + 255) / 256;
    cn_init_kernel<<<blocks, 256, 0, stream>>>(out, n, counters);
    cn_bucket_kernel<<<blocks, 256, 0, stream>>>(species, n, counters, lists, n);

    dim3 grid((n + TM - 1) / TM, NS);
    cn_mlp_kernel<<<grid, 256, 0, stream>>>(emb, W1, b1, W2, b2, W3, b3,
                                            counters, lists, n, out);
}